// LeadingZeroDetector48_43860206027314
// MI455X (gfx1250) — compile-verified
//
#include <hip/hip_runtime.h>
#include <stdint.h>

// LZC of 48-bit words (floats 0/1) -> 6-bit binary LZC as floats.
// Memory-bound: ~453 MB total traffic -> ~19.5 us floor @ 23.3 TB/s.
// Data path: async DMA global->LDS (padded scatter), per-lane bit-pack + clz,
// async DMA LDS->global for the output tile.

#define NTHREADS        256
#define WORDS_PER_BLOCK 256
#define WORD_DW         48   // dwords per word in global memory
#define PAD_DW          52   // dwords per word in LDS (208 B, 16B-aligned, 2-way banked)
#define OUT_F           6

__global__ __launch_bounds__(NTHREADS)
void lzc48_kernel(const float* __restrict__ X, float* __restrict__ out, unsigned nwords) {
    __shared__ __align__(16) float lin[WORDS_PER_BLOCK * PAD_DW];   // 53248 B
    __shared__ __align__(16) float lout[WORDS_PER_BLOCK * OUT_F];   // 6144 B

    const unsigned t  = threadIdx.x;
    const unsigned w0 = blockIdx.x * WORDS_PER_BLOCK;
    if (w0 >= nwords) return;  // whole block uniform

    // ---------------- Stage 1: global -> LDS (async DMA, coalesced) ----------------
    // 256 words * 12 granules(16B) = 3072 granules; consecutive lanes take
    // consecutive granules (perfect b128 coalescing). LDS destination is a
    // padded scatter: granule g of word w=g/12 lands at w*208 + (g%12)*16
    //               = 16*(g + g/12) bytes.
    const uint64_t in_base  = (uint64_t)(uintptr_t)(X + (size_t)w0 * WORD_DW);
    const uint32_t lin_base = (uint32_t)(uintptr_t)(&lin[0]);  // low 32 bits of flat shared ptr == LDS offset

#pragma unroll
    for (int it = 0; it < 12; ++it) {
        uint32_t g    = (uint32_t)it * NTHREADS + t;          // 0..3071
        uint32_t q    = g / 12u;                              // word index (magic-mul)
        uint32_t goff = g << 4;                               // global byte offset
        uint32_t loff = lin_base + ((g + q) << 4);            // padded LDS byte offset
        asm volatile("global_load_async_to_lds_b128 %0, %1, %2"
                     :
                     : "v"(loff), "v"(goff), "s"(in_base)
                     : "memory");
    }
    asm volatile("s_wait_asynccnt 0x0" ::: "memory");
    __syncthreads();

    // ---------------- Stage 2: per-lane LZC ----------------
    // Bit i of the word: X[i] in {0.0f, 1.0f}; 1.0f = 0x3F800000 -> bit29 set.
    // m0 holds X[0..31] with X[0] at bit31; m1 holds X[32..47] at bits 31..16.
    {
        const float4* wp = (const float4*)&lin[t * PAD_DW];
        uint32_t m0 = 0u, m1 = 0u;
#pragma unroll
        for (int j = 0; j < 12; ++j) {
            float4 v = wp[j];
            uint32_t b0 = (__float_as_uint(v.x) >> 29) & 1u;
            uint32_t b1 = (__float_as_uint(v.y) >> 29) & 1u;
            uint32_t b2 = (__float_as_uint(v.z) >> 29) & 1u;
            uint32_t b3 = (__float_as_uint(v.w) >> 29) & 1u;
            const int i = j * 4;
            if (i < 32) {
                m0 |= (b0 << (31 - i)) | (b1 << (30 - i)) | (b2 << (29 - i)) | (b3 << (28 - i));
            } else {
                m1 |= (b0 << (63 - i)) | (b1 << (62 - i)) | (b2 << (61 - i)) | (b3 << (60 - i));
            }
        }
        uint32_t lzc = m0 ? (uint32_t)__builtin_clz(m0)
                          : (m1 ? 32u + (uint32_t)__builtin_clz(m1) : 48u);
        // Expand 6-bit LZC, MSB first, into the LDS output tile.
        float* op = &lout[t * OUT_F];
#pragma unroll
        for (int j = 0; j < 6; ++j)
            op[j] = (float)((lzc >> (5 - j)) & 1u);
    }
    __syncthreads();

    // ---------------- Stage 3: LDS -> global (async DMA, coalesced) ----------------
    // 256 words * 24 B = 6144 B = 384 granules of 16 B.
    const uint64_t out_base  = (uint64_t)(uintptr_t)(out + (size_t)w0 * OUT_F);
    const uint32_t lout_base = (uint32_t)(uintptr_t)(&lout[0]);
    {
        uint32_t goff = t << 4;
        uint32_t loff = lout_base + (t << 4);
        asm volatile("global_store_async_from_lds_b128 %0, %1, %2"
                     :
                     : "v"(goff), "v"(loff), "s"(out_base)
                     : "memory");
    }
    if (t < 128u) {
        uint32_t g    = 256u + t;
        uint32_t goff = g << 4;
        uint32_t loff = lout_base + (g << 4);
        asm volatile("global_store_async_from_lds_b128 %0, %1, %2"
                     :
                     : "v"(goff), "v"(loff), "s"(out_base)
                     : "memory");
    }
    asm volatile("s_wait_asynccnt 0x0" ::: "memory");  // S_ENDPGM also implies wait-idle
}

extern "C" void kernel_launch(void* const* d_in, const int* in_sizes, int n_in,
                              void* d_out, int out_size, void* d_ws, size_t ws_size,
                              hipStream_t stream) {
    const float* X  = (const float*)d_in[0];
    float* out      = (float*)d_out;
    const unsigned nwords = (unsigned)(in_sizes[0] / WORD_DW);   // B = 2^21
    const unsigned nblocks = (nwords + WORDS_PER_BLOCK - 1) / WORDS_PER_BLOCK;  // 8192, exact
    lzc48_kernel<<<nblocks, NTHREADS, 0, stream>>>(X, out, nwords);
}